// SqueezeExciteBlock_5970004542268
// MI455X (gfx1250) — compile-verified
//
#include <hip/hip_runtime.h>
#include <hip/hip_bf16.h>
#include <math.h>

// Problem constants (match setup_inputs): x[B,C,L], w1[R,C], b1[R], w2[C,R], b2[C]
#define SE_B 64
#define SE_C 256
#define SE_L 4096
#define SE_R 16
#define SE_ROWS (SE_B * SE_C)          // 16384 (b,c) rows
#define SE_L4 (SE_L / 4)               // 1024 float4 per row

typedef __attribute__((ext_vector_type(2))) float v2f;
typedef __attribute__((ext_vector_type(4))) float v4f;
typedef __attribute__((ext_vector_type(8))) float v8f;

// ---------------------------------------------------------------------------
// Kernel 1: mean over L for each (b,c) row.  Pure streaming read: 256 MB.
// ---------------------------------------------------------------------------
__global__ __launch_bounds__(256) void se_mean_kernel(const float* __restrict__ x,
                                                      float* __restrict__ mean) {
    const int row = blockIdx.x;                       // row = b*C + c
    const v4f* __restrict__ xr = (const v4f*)(x + (size_t)row * SE_L);

    float s = 0.0f;
    #pragma unroll
    for (int i = 0; i < SE_L4 / 256; ++i) {           // 4 iters of b128 loads
        v4f v = xr[threadIdx.x + i * 256];
        s += (v.x + v.y) + (v.z + v.w);
    }

    // wave32 reduction
    #pragma unroll
    for (int off = 16; off > 0; off >>= 1)
        s += __shfl_down(s, off, 32);

    __shared__ float red[8];                          // 8 waves per block
    const int lane = threadIdx.x & 31;
    const int wv   = threadIdx.x >> 5;
    if (lane == 0) red[wv] = s;
    __syncthreads();

    if (threadIdx.x == 0) {
        float t = 0.0f;
        #pragma unroll
        for (int i = 0; i < 8; ++i) t += red[i];
        mean[row] = t * (1.0f / (float)SE_L);
    }
}

// ---------------------------------------------------------------------------
// Kernel 2: excitation MLP on the matrix pipe (V_WMMA_F32_16X16X4_F32).
//   y1 = relu(mean @ w1^T + b1)        [64 x 16]
//   gate = sigmoid(y1 @ w2^T + b2)     [64 x 256]
// One block, 4 waves (wave32).  Wave w owns M-tile rows [16w, 16w+16).
// Fragment layouts per CDNA5 ISA 7.12.2:
//   A 16x4 f32 : lane m=lane%16; {a.x,a.y} = A[m][k0+2*(lane/16) + {0,1}]
//   B 4x16 f32 : lane n=lane%16; {b.x,b.y} = B[k0+2*(lane/16) + {0,1}][n]
//   C/D 16x16  : vgpr i -> (m = i + 8*(lane/16), n = lane%16)
// ---------------------------------------------------------------------------
__global__ __launch_bounds__(128) void se_mlp_kernel(const float* __restrict__ mean,
                                                     const float* __restrict__ w1,
                                                     const float* __restrict__ b1,
                                                     const float* __restrict__ w2,
                                                     const float* __restrict__ b2,
                                                     float* __restrict__ gate) {
    __shared__ float y1s[SE_B * SE_R];                // 64x16 = 4 KB

    const int lane  = threadIdx.x & 31;
    const int wv    = threadIdx.x >> 5;               // 0..3 -> M tile
    const int mrow  = wv * 16 + (lane & 15);          // A-fragment row
    const int khalf = lane >> 4;                      // 0: K={0,1}, 1: K={2,3}
    const int ncol  = lane & 15;                      // B/C-fragment column

    // ---- GEMM1: [64x256] x [256x16] -> [64x16], K = 256 (64 WMMA steps) ----
    v8f acc = {};
    for (int k0 = 0; k0 < SE_C; k0 += 4) {
        const int ka = k0 + 2 * khalf;
        v2f a, b;
        a.x = mean[mrow * SE_C + ka];
        a.y = mean[mrow * SE_C + ka + 1];
        // B[k][n] = w1[n][k]  (w1 is [R,C] row-major)
        b.x = w1[ncol * SE_C + ka];
        b.y = w1[ncol * SE_C + ka + 1];
        acc = __builtin_amdgcn_wmma_f32_16x16x4_f32(
            false, a, false, b, (short)0, acc, false, false);
    }
    // bias + ReLU, stage y1 in LDS
    {
        const float bias = b1[ncol];
        #pragma unroll
        for (int i = 0; i < 8; ++i) {
            const int m = wv * 16 + i + 8 * khalf;
            float v = acc[i] + bias;
            y1s[m * SE_R + ncol] = v > 0.0f ? v : 0.0f;
        }
    }
    __syncthreads();

    // ---- GEMM2: [64x16] x [16x256] -> [64x256], K = 16 (4 WMMA steps/tile) --
    for (int nt = 0; nt < SE_C / 16; ++nt) {          // 16 N-tiles per wave
        v8f acc2 = {};
        const int nglob = nt * 16 + ncol;
        #pragma unroll
        for (int k0 = 0; k0 < SE_R; k0 += 4) {
            const int ka = k0 + 2 * khalf;
            v2f a, b;
            a.x = y1s[mrow * SE_R + ka];
            a.y = y1s[mrow * SE_R + ka + 1];
            // B[k][n] = w2[n][k]  (w2 is [C,R] row-major)
            b.x = w2[nglob * SE_R + ka];
            b.y = w2[nglob * SE_R + ka + 1];
            acc2 = __builtin_amdgcn_wmma_f32_16x16x4_f32(
                false, a, false, b, (short)0, acc2, false, false);
        }
        const float bias = b2[nglob];
        #pragma unroll
        for (int i = 0; i < 8; ++i) {
            const int m = wv * 16 + i + 8 * khalf;
            float v = acc2[i] + bias;
            gate[m * SE_C + nglob] = 1.0f / (1.0f + __expf(-v));
        }
    }
}

// ---------------------------------------------------------------------------
// Kernel 3: out = x * gate[b,c].  Streaming 256 MB in, 256 MB out.
// Gate index is block-uniform -> scalar load.  NT loads (x is dead after this
// pass) and NT stores (output never re-read) keep L2 clean.
// ---------------------------------------------------------------------------
__global__ __launch_bounds__(256) void se_scale_kernel(const float* __restrict__ x,
                                                       const float* __restrict__ gate,
                                                       float* __restrict__ out) {
    const int row = blockIdx.x;
    const float g = gate[row];
    const v4f* __restrict__ xr = (const v4f*)(x + (size_t)row * SE_L);
    v4f* __restrict__ orow     = (v4f*)(out + (size_t)row * SE_L);

    #pragma unroll
    for (int i = 0; i < SE_L4 / 256; ++i) {
        const int idx = threadIdx.x + i * 256;
        v4f v = __builtin_nontemporal_load(&xr[idx]);
        v *= g;
        __builtin_nontemporal_store(v, &orow[idx]);
    }
}

// ---------------------------------------------------------------------------
extern "C" void kernel_launch(void* const* d_in, const int* in_sizes, int n_in,
                              void* d_out, int out_size, void* d_ws, size_t ws_size,
                              hipStream_t stream) {
    const float* x  = (const float*)d_in[0];   // [B,C,L]
    const float* w1 = (const float*)d_in[1];   // [R,C]
    const float* b1 = (const float*)d_in[2];   // [R]
    const float* w2 = (const float*)d_in[3];   // [C,R]
    const float* b2 = (const float*)d_in[4];   // [C]
    float* out = (float*)d_out;

    float* mean = (float*)d_ws;                // [B*C]
    float* gate = mean + SE_ROWS;              // [B*C]

    se_mean_kernel <<<SE_ROWS, 256, 0, stream>>>(x, mean);
    se_mlp_kernel  <<<1,       128, 0, stream>>>(mean, w1, b1, w2, b2, gate);
    se_scale_kernel<<<SE_ROWS, 256, 0, stream>>>(x, gate, out);
}